// RGAT_5351529251343
// MI455X (gfx1250) — compile-verified
//
#include <hip/hip_runtime.h>
#include <hip/hip_bf16.h>

// ---------------------------------------------------------------------------
// Problem constants (fixed by the reference)
// ---------------------------------------------------------------------------
#define NN0 30000
#define NN1 20000
#define NNODE 50000
#define NEDGE 400000
#define DHID 64
#define NCLS 16
#define INF0 128

typedef float v2f __attribute__((ext_vector_type(2)));
typedef float v8f __attribute__((ext_vector_type(8)));

static inline int cdiv(long a, long b) { return (int)((a + b - 1) / b); }

// ---------------------------------------------------------------------------
// fp32 WMMA GEMM:  C[r, m] = sum_k A[r, k] * B[k, m] (+ bias[m])
//
// Block = 5 waves (160 threads); wave w owns row strip (blockIdx*5+w)*16.
// All row counts (3125 / 1875 / 1250 strips) divide by 5 -> exact grids,
// no guards, EXEC all-ones for WMMA legality.
//
// B panel is staged in LDS *pair-swizzled*: element (k, c) lives at
//   Bs[(k>>1)*CW*2 + c*2 + (k&1)]
// so a wave's B fragment (rows k0, k0+1 of one column; k0 always even) is a
// single aligned ds_load_b64 straight into an even VGPR pair -> no repacking
// movs in front of v_wmma_f32_16x16x4_f32.
//
// VGPR layouts per CDNA5 ISA 7.12.2:
//   A 16x4 : lanes 0-15 hold K={0,1}; lanes 16-31 hold K={2,3}
//   B 4x16 : lanes 0-15 hold rows K={0,1}; lanes 16-31 rows K={2,3}
//   D 16x16: VGPR v -> M=v (lanes 0-15) / M=v+8 (lanes 16-31), N = lane&15
// ---------------------------------------------------------------------------
template <int NT>
__global__ void gemm_wmma_f32_t(const float* __restrict__ A, int lda,
                                const float* __restrict__ B, int ldb,
                                float* __restrict__ C, int ldc,
                                const float* __restrict__ bias,
                                int K, int M) {
  constexpr int CW = NT * 16;            // panel width
  __shared__ float Bs[256 * CW];         // K <= 256 (pair-swizzled layout)

  const int lane = threadIdx.x & 31;
  const int wave = threadIdx.x >> 5;     // 0..4
  const int ml   = lane & 15;
  const int kh   = lane >> 4;            // K-half select
  const long rbase = ((long)blockIdx.x * 5 + wave) * 16;
  const float* Arow = A + (rbase + ml) * (long)lda;

  for (int nb = 0; nb < M; nb += CW) {
    __syncthreads();  // protect LDS WAR vs previous panel
    for (int t = threadIdx.x; t < K * CW; t += blockDim.x) {
      const int k = t / CW, c = t % CW;
      Bs[(k >> 1) * CW * 2 + c * 2 + (k & 1)] = B[(long)k * ldb + nb + c];
    }
    __syncthreads();

    v8f acc[NT] = {};
    for (int kb = 0; kb < K; kb += 4) {
      const int k0 = kb + 2 * kh;        // even
      v2f a;
      a.x = Arow[k0];
      a.y = Arow[k0 + 1];
      const float* brow = Bs + (k0 >> 1) * CW * 2 + ml * 2;
#pragma unroll
      for (int j = 0; j < NT; ++j) {
        const v2f b = *(const v2f*)(brow + j * 32);
        acc[j] = __builtin_amdgcn_wmma_f32_16x16x4_f32(
            /*neg_a=*/false, a, /*neg_b=*/false, b,
            /*c_mod=*/(short)0, acc[j], /*reuse_a=*/false, /*reuse_b=*/false);
      }
    }

#pragma unroll
    for (int j = 0; j < NT; ++j) {
      const float bv = bias ? bias[nb + j * 16 + ml] : 0.0f;
#pragma unroll
      for (int v = 0; v < 8; ++v) {
        C[(rbase + v + 8 * kh) * (long)ldc + nb + j * 16 + ml] = acc[j][v] + bv;
      }
    }
  }
}

static inline void launch_gemm(const float* A, int lda, const float* B, int ldb,
                               float* C, int ldc, const float* bias,
                               int rows, int K, int M, hipStream_t stream) {
  const int strips = rows / 16;          // exact (all row counts % 16 == 0)
  const int blocks = strips / 5;         // exact (all strip counts % 5 == 0)
  if (M % 64 == 0) {
    gemm_wmma_f32_t<4><<<blocks, 160, 0, stream>>>(A, lda, B, ldb, C, ldc, bias, K, M);
  } else {
    gemm_wmma_f32_t<1><<<blocks, 160, 0, stream>>>(A, lda, B, ldb, C, ldc, bias, K, M);
  }
}

// ---------------------------------------------------------------------------
// Small helper kernels
// ---------------------------------------------------------------------------
__global__ void fillf(float* __restrict__ p, float v, long n) {
  long t = (long)blockIdx.x * blockDim.x + threadIdx.x;
  if (t < n) p[t] = v;
}

// softmax of mix_w (2 types x 3 layers x 2 graphs) over last axis
__global__ void mix_softmax(const float* __restrict__ mw, float* __restrict__ w) {
  if (threadIdx.x == 0 && blockIdx.x == 0) {
    for (int p = 0; p < 6; ++p) {
      float a = mw[2 * p], b = mw[2 * p + 1];
      float m = fmaxf(a, b);
      float ea = __expf(a - m), eb = __expf(b - m);
      float s = ea + eb;
      w[2 * p] = ea / s;
      w[2 * p + 1] = eb / s;
    }
  }
}

// el[n,h] = sum_d feat[n,h,d]*al[h,d] ; er analogous
__global__ void compute_elr(const float* __restrict__ feat,
                            const float* __restrict__ al,
                            const float* __restrict__ ar,
                            float* __restrict__ el, float* __restrict__ er,
                            int H, int D, int nN) {
  int t = blockIdx.x * blockDim.x + threadIdx.x;
  if (t >= nN * H) return;
  int n = t / H, h = t % H;
  const float* f = feat + (long)n * H * D + (long)h * D;
  const float* a = al + h * D;
  const float* b = ar + h * D;
  float sl = 0.f, sr = 0.f;
  for (int d = 0; d < D; ++d) {
    float v = f[d];
    sl += v * a[d];
    sr += v * b[d];
  }
  el[t] = sl;
  er[t] = sr;
}

__device__ inline void atomicMaxF(float* addr, float val) {
  unsigned int* ua = (unsigned int*)addr;
  unsigned int old = *ua;
  while (__uint_as_float(old) < val) {
    unsigned int assumed = old;
    old = atomicCAS(ua, assumed, __float_as_uint(val));
    if (old == assumed) break;
  }
}

// pass A: e = leaky_relu(el[src]+er[dst]); store e; segment-max into mnode[dst]
__global__ void edge_logits(const int* __restrict__ src, const int* __restrict__ dst,
                            const float* __restrict__ el, const float* __restrict__ er,
                            float* __restrict__ eb, float* __restrict__ mnode,
                            int H, int nE) {
  int e = blockIdx.x * blockDim.x + threadIdx.x;
  if (e >= nE) return;
  int s = src[e], d = dst[e];
  for (int h = 0; h < H; ++h) {
    float v = el[(long)s * H + h] + er[(long)d * H + h];
    v = v > 0.f ? v : 0.2f * v;  // SLOPE = 0.2
    eb[(long)e * H + h] = v;
    atomicMaxF(&mnode[(long)d * H + h], v);
  }
}

// pass B: ex = exp(e - m[dst]); store ex; segment-sum into den[dst]
__global__ void edge_exp(const int* __restrict__ dst, float* __restrict__ eb,
                         const float* __restrict__ mnode, float* __restrict__ den,
                         int H, int nE) {
  int e = blockIdx.x * blockDim.x + threadIdx.x;
  if (e >= nE) return;
  int d = dst[e];
  for (int h = 0; h < H; ++h) {
    float ex = __expf(eb[(long)e * H + h] - mnode[(long)d * H + h]);
    eb[(long)e * H + h] = ex;
    atomicAdd(&den[(long)d * H + h], ex);
  }
}

// pass C: out[dst] += feat[src] * (ex/den[dst]) ; one thread per (edge, 4 cols)
__global__ void edge_aggregate(const int* __restrict__ src, const int* __restrict__ dst,
                               const float* __restrict__ feat, const float* __restrict__ eb,
                               const float* __restrict__ den, float* __restrict__ outb,
                               int H, int D, int M, int nE) {
  long t = (long)blockIdx.x * blockDim.x + threadIdx.x;
  const int chunks = M >> 2;
  if (t >= (long)nE * chunks) return;
  int e = (int)(t / chunks);
  int j = (int)(t % chunks) << 2;
  int h = j / D;
  int s = src[e], d = dst[e];
  float alpha = eb[(long)e * H + h] / den[(long)d * H + h];
  const float4 f = *(const float4*)(feat + (long)s * M + j);
  float* o = outb + (long)d * M + j;
  atomicAdd(o + 0, f.x * alpha);
  atomicAdd(o + 1, f.y * alpha);
  atomicAdd(o + 2, f.z * alpha);
  atomicAdd(o + 3, f.w * alpha);
}

// layers 0/1: acc += elu(out (+resid) + bias) * w[type, layer, graph]
__global__ void finalize01(float* __restrict__ accb, const float* __restrict__ outb,
                           const float* __restrict__ resid, const float* __restrict__ bias,
                           const float* __restrict__ wmix, int layer, int graph,
                           int M, int nN) {
  long t = (long)blockIdx.x * blockDim.x + threadIdx.x;
  if (t >= (long)nN * M) return;
  int n = (int)(t / M);
  int j = (int)(t % M);
  float v = outb[t] + bias[j];
  if (resid) v += resid[t];
  v = v > 0.f ? v : (__expf(v) - 1.f);  // elu
  int ty = (n >= NN0) ? 1 : 0;
  accb[t] += v * wmix[ty * 6 + layer * 2 + graph];
}

// layer 2: logits += (out + res_fc + bias) * w[type, 2, graph]   (no act, H2=1)
__global__ void finalize2(float* __restrict__ logits, const float* __restrict__ outb,
                          const float* __restrict__ resb, const float* __restrict__ bias,
                          const float* __restrict__ wmix, int graph, int nN) {
  long t = (long)blockIdx.x * blockDim.x + threadIdx.x;
  if (t >= (long)nN * NCLS) return;
  int n = (int)(t / NCLS);
  int j = (int)(t % NCLS);
  float v = outb[t] + resb[t] + bias[j];
  int ty = (n >= NN0) ? 1 : 0;
  logits[t] += v * wmix[ty * 6 + 4 + graph];
}

// ---------------------------------------------------------------------------
// Launcher
// ---------------------------------------------------------------------------
extern "C" void kernel_launch(void* const* d_in, const int* in_sizes, int n_in,
                              void* d_out, int out_size, void* d_ws, size_t ws_size,
                              hipStream_t stream) {
  const float* features0 = (const float*)d_in[0];
  const float* features1 = (const float*)d_in[1];
  const float* fc_w0 = (const float*)d_in[2];
  const float* fc_b0 = (const float*)d_in[3];
  const float* fc_w1 = (const float*)d_in[4];
  const float* fc_b1 = (const float*)d_in[5];
  const float* mix_w = (const float*)d_in[6];
  const float* W0  = (const float*)d_in[7];
  const float* al0 = (const float*)d_in[8];
  const float* ar0 = (const float*)d_in[9];
  const float* b0  = (const float*)d_in[10];
  const float* W1  = (const float*)d_in[11];
  const float* al1 = (const float*)d_in[12];
  const float* ar1 = (const float*)d_in[13];
  const float* b1  = (const float*)d_in[14];
  const float* W2  = (const float*)d_in[15];
  const float* al2 = (const float*)d_in[16];
  const float* ar2 = (const float*)d_in[17];
  const float* b2  = (const float*)d_in[18];
  const float* res2 = (const float*)d_in[19];
  const int* srcs[2] = {(const int*)d_in[20], (const int*)d_in[22]};
  const int* dsts[2] = {(const int*)d_in[21], (const int*)d_in[23]};

  // workspace layout (floats)
  float* ws = (float*)d_ws;
  float* h0   = ws;                                // N x 64
  float* bufA = h0 + (long)NNODE * 64;             // N x 256
  float* bufB = bufA + (long)NNODE * 256;          // N x 256
  float* feat = bufB + (long)NNODE * 256;          // N x 256
  float* outb = feat + (long)NNODE * 256;          // N x 256
  float* resb = outb + (long)NNODE * 256;          // N x 16
  float* el   = resb + (long)NNODE * 16;           // N x 4
  float* er   = el + (long)NNODE * 4;              // N x 4
  float* mn   = er + (long)NNODE * 4;              // N x 4
  float* dn   = mn + (long)NNODE * 4;              // N x 4
  float* eb   = dn + (long)NNODE * 4;              // E x 4
  float* wmix = eb + (long)NEDGE * 4;              // 12

  const int TB = 256;

  // mix weights
  mix_softmax<<<1, 32, 0, stream>>>(mix_w, wmix);

  // input projections: h0 = [f0 @ fc_w0 + b0 ; f1 @ fc_w1 + b1]
  launch_gemm(features0, INF0, fc_w0, DHID, h0, DHID, fc_b0, NN0, INF0, DHID, stream);
  launch_gemm(features1, INF0, fc_w1, DHID, h0 + (long)NN0 * DHID, DHID, fc_b1,
              NN1, INF0, DHID, stream);

  // ---------------- layers 0 and 1 (H=4, D=64, M=256) ----------------
  struct Lyr {
    const float* h; int K; const float *W, *al, *ar, *b, *resid; float* acc;
  };
  Lyr layers[2] = {
      {h0,   DHID, W0, al0, ar0, b0, nullptr, bufA},
      {bufA, 256,  W1, al1, ar1, b1, bufA,    bufB},
  };

  for (int l = 0; l < 2; ++l) {
    const Lyr& L = layers[l];
    const int H = 4, D = 64, M = 256;
    fillf<<<cdiv((long)NNODE * M, TB), TB, 0, stream>>>(L.acc, 0.f, (long)NNODE * M);
    for (int i = 0; i < 2; ++i) {
      launch_gemm(L.h, L.K, L.W + (long)i * L.K * M, M, feat, M, nullptr,
                  NNODE, L.K, M, stream);
      compute_elr<<<cdiv((long)NNODE * H, TB), TB, 0, stream>>>(
          feat, L.al + (long)i * H * D, L.ar + (long)i * H * D, el, er, H, D, NNODE);
      fillf<<<cdiv((long)NNODE * H, TB), TB, 0, stream>>>(mn, -1e30f, (long)NNODE * H);
      fillf<<<cdiv((long)NNODE * H, TB), TB, 0, stream>>>(dn, 0.f, (long)NNODE * H);
      fillf<<<cdiv((long)NNODE * M, TB), TB, 0, stream>>>(outb, 0.f, (long)NNODE * M);
      edge_logits<<<cdiv(NEDGE, TB), TB, 0, stream>>>(srcs[i], dsts[i], el, er, eb, mn, H, NEDGE);
      edge_exp<<<cdiv(NEDGE, TB), TB, 0, stream>>>(dsts[i], eb, mn, dn, H, NEDGE);
      edge_aggregate<<<cdiv((long)NEDGE * (M / 4), TB), TB, 0, stream>>>(
          srcs[i], dsts[i], feat, eb, dn, outb, H, D, M, NEDGE);
      finalize01<<<cdiv((long)NNODE * M, TB), TB, 0, stream>>>(
          L.acc, outb, L.resid, L.b + (long)i * M, wmix, l, i, M, NNODE);
    }
  }

  // ---------------- layer 2 (H=1, D=NC=16, res_fc) ----------------
  float* logits = (float*)d_out;
  fillf<<<cdiv((long)NNODE * NCLS, TB), TB, 0, stream>>>(logits, 0.f, (long)NNODE * NCLS);
  for (int i = 0; i < 2; ++i) {
    const int H = 1, D = NCLS, M = NCLS, K = 256;
    launch_gemm(bufB, K, W2 + (long)i * K * M, M, feat, M, nullptr, NNODE, K, M, stream);
    launch_gemm(bufB, K, res2 + (long)i * K * M, M, resb, M, nullptr, NNODE, K, M, stream);
    compute_elr<<<cdiv((long)NNODE * H, TB), TB, 0, stream>>>(
        feat, al2 + (long)i * H * D, ar2 + (long)i * H * D, el, er, H, D, NNODE);
    fillf<<<cdiv((long)NNODE * H, TB), TB, 0, stream>>>(mn, -1e30f, (long)NNODE * H);
    fillf<<<cdiv((long)NNODE * H, TB), TB, 0, stream>>>(dn, 0.f, (long)NNODE * H);
    fillf<<<cdiv((long)NNODE * M, TB), TB, 0, stream>>>(outb, 0.f, (long)NNODE * M);
    edge_logits<<<cdiv(NEDGE, TB), TB, 0, stream>>>(srcs[i], dsts[i], el, er, eb, mn, H, NEDGE);
    edge_exp<<<cdiv(NEDGE, TB), TB, 0, stream>>>(dsts[i], eb, mn, dn, H, NEDGE);
    edge_aggregate<<<cdiv((long)NEDGE * (M / 4), TB), TB, 0, stream>>>(
        srcs[i], dsts[i], feat, eb, dn, outb, H, D, M, NEDGE);
    finalize2<<<cdiv((long)NNODE * NCLS, TB), TB, 0, stream>>>(
        logits, outb, resb, b2 + (long)i * NCLS, wmix, i, NNODE);
  }
}